// CausalSelfAttention_55293408968728
// MI455X (gfx1250) — compile-verified
//
#include <hip/hip_runtime.h>
#include <hip/hip_bf16.h>

typedef __attribute__((ext_vector_type(2))) float v2f;
typedef __attribute__((ext_vector_type(8))) float v8f;

#define B_SZ 2
#define T_SZ 2048
#define C_SZ 1024
#define H_N 16
#define HD_SZ 64

__device__ __forceinline__ v8f wmma4(v2f a, v2f b, v8f c) {
  // D = A(16x4) * B(4x16) + C, fp32 tensor op
  return __builtin_amdgcn_wmma_f32_16x16x4_f32(false, a, false, b, (short)0, c,
                                               false, false);
}

// ---------------------------------------------------------------------------
// 32x32 LDS-tiled transpose: D[c,r] = S[r,c].  S is [R, Cc] row-major.
// ---------------------------------------------------------------------------
__global__ __launch_bounds__(256) void transpose_mat(
    const float* __restrict__ S, float* __restrict__ D, int R, int Cc) {
  __shared__ float tile[32][33];
  const int ctiles = Cc >> 5;
  const int c0 = (blockIdx.x % ctiles) << 5;
  const int r0 = (blockIdx.x / ctiles) << 5;
  const int tx = threadIdx.x & 31;
  const int ty = threadIdx.x >> 5;  // 0..7
#pragma unroll
  for (int i = 0; i < 32; i += 8)
    tile[ty + i][tx] = S[(size_t)(r0 + ty + i) * Cc + c0 + tx];
  __syncthreads();
#pragma unroll
  for (int i = 0; i < 32; i += 8)
    D[(size_t)(c0 + ty + i) * R + r0 + tx] = tile[tx][ty + i];
}

// ---------------------------------------------------------------------------
// GEMM: C[M,N] = A[M,K] @ B[K,N] + bias[N], with B supplied TRANSPOSED
// (Bt[N,K] row-major) so both fragments are contiguous b64 loads.
// One wave computes a 64x64 tile: 8 b64 loads : 16 wmma per k-step.
// ---------------------------------------------------------------------------
__global__ __launch_bounds__(32) void gemm_bias_wmma(
    const float* __restrict__ A, const float* __restrict__ Bt,
    const float* __restrict__ bias, float* __restrict__ C, int M, int N,
    int K) {
  const int lane = threadIdx.x;
  const int r = lane & 15;   // M-row / N-col within 16
  const int hh = lane >> 4;  // which K-pair half
  const int ntiles = N >> 6;
  const int m0 = (blockIdx.x / ntiles) << 6;
  const int n0 = (blockIdx.x % ntiles) << 6;

  v8f acc[4][4];
#pragma unroll
  for (int i = 0; i < 4; ++i)
#pragma unroll
    for (int j = 0; j < 4; ++j)
#pragma unroll
      for (int v = 0; v < 8; ++v) acc[i][j][v] = 0.0f;

  const float* Ap0 = A + (size_t)(m0 + r) * K + 2 * hh;
  const float* Bp0 = Bt + (size_t)(n0 + r) * K + 2 * hh;

  for (int k0 = 0; k0 < K; k0 += 4) {
    v2f a[4], b[4];
#pragma unroll
    for (int i = 0; i < 4; ++i)
      a[i] = *(const v2f*)(Ap0 + (size_t)i * 16 * K + k0);
#pragma unroll
    for (int j = 0; j < 4; ++j)
      b[j] = *(const v2f*)(Bp0 + (size_t)j * 16 * K + k0);
#pragma unroll
    for (int j = 0; j < 4; ++j)
#pragma unroll
      for (int i = 0; i < 4; ++i) acc[i][j] = wmma4(a[i], b[j], acc[i][j]);
  }

#pragma unroll
  for (int j = 0; j < 4; ++j) {
    const float bj = bias[n0 + 16 * j + r];
#pragma unroll
    for (int i = 0; i < 4; ++i) {
      float* Cp = C + (size_t)(m0 + 16 * i + 8 * hh) * N + n0 + 16 * j + r;
#pragma unroll
      for (int v = 0; v < 8; ++v) Cp[(size_t)v * N] = acc[i][j][v] + bj;
    }
  }
}

// ---------------------------------------------------------------------------
// RoPE on Q,K + reorg: Qr,Kr -> [BH, T, 64] row-major; V -> Vt [BH, 64, T].
// One 32-lane group per (bh, t); lane d = 0..31 handles the (d, d+32) pair.
// ---------------------------------------------------------------------------
__global__ __launch_bounds__(256) void rope_reorg(const float* __restrict__ qkv,
                                                  float* __restrict__ Qr,
                                                  float* __restrict__ Kr,
                                                  float* __restrict__ Vt) {
  const int unit = blockIdx.x * 8 + (threadIdx.x >> 5);
  const int d = threadIdx.x & 31;
  const int bh = unit >> 11;  // / T_SZ
  const int t = unit & (T_SZ - 1);
  const int b = bh >> 4;
  const int head = bh & 15;

  const float* base =
      qkv + (size_t)(b * T_SZ + t) * (3 * C_SZ) + head * HD_SZ;
  const float q1 = base[d], q2 = base[d + 32];
  const float k1 = base[C_SZ + d], k2 = base[C_SZ + d + 32];
  const float v1 = base[2 * C_SZ + d], v2 = base[2 * C_SZ + d + 32];

  const float ts = powf(10000.0f, (float)d * (1.0f / 32.0f));
  const float ang = (float)t / ts;
  const float sn = sinf(ang);
  const float cs = cosf(ang);

  const size_t qk = ((size_t)bh * T_SZ + t) * HD_SZ;
  Qr[qk + d] = q1 * cs - q2 * sn;
  Qr[qk + d + 32] = q2 * cs + q1 * sn;
  Kr[qk + d] = k1 * cs - k2 * sn;
  Kr[qk + d + 32] = k2 * cs + k1 * sn;

  const size_t vb = ((size_t)bh * HD_SZ + d) * T_SZ + t;
  Vt[vb] = v1;
  Vt[vb + (size_t)32 * T_SZ] = v2;
}

// ---------------------------------------------------------------------------
// Flash attention: one wave per (bh, 16-row q tile), streaming 64-key blocks.
// ---------------------------------------------------------------------------
__global__ __launch_bounds__(32) void flash_attn(const float* __restrict__ Qr,
                                                 const float* __restrict__ Kr,
                                                 const float* __restrict__ Vt,
                                                 float* __restrict__ Y) {
  __shared__ float Pl[16 * 64];  // P tile bounce buffer (C-layout -> A-layout)
  const int lane = threadIdx.x;
  const int r = lane & 15;
  const int hh = lane >> 4;
  const int qi = blockIdx.x & 127;  // q tile index (T/16 = 128)
  const int bh = blockIdx.x >> 7;
  const int q0 = qi << 4;

  // Q fragment in A-layout registers: aq[s] covers k' = 4s + 2h .. +1
  v2f aq[16];
  {
    const float* Qp = Qr + ((size_t)bh * T_SZ + q0 + r) * HD_SZ + 2 * hh;
#pragma unroll
    for (int s = 0; s < 16; ++s) aq[s] = *(const v2f*)(Qp + 4 * s);
  }

  v8f o[4];
  float mrow[8], lrow[8];
#pragma unroll
  for (int jn = 0; jn < 4; ++jn)
#pragma unroll
    for (int v = 0; v < 8; ++v) o[jn][v] = 0.0f;
#pragma unroll
  for (int v = 0; v < 8; ++v) {
    mrow[v] = -1.0e30f;
    lrow[v] = 0.0f;
  }

  const int jmax = qi >> 2;  // last key block index (64*jmax <= q0)
  for (int j = 0; j <= jmax; ++j) {
    const int kbase = j << 6;

    // ---- S = Q @ K^T (16 x 64) ----
    v8f c[4];
#pragma unroll
    for (int jn = 0; jn < 4; ++jn)
#pragma unroll
      for (int v = 0; v < 8; ++v) c[jn][v] = 0.0f;

    const float* Kp = Kr + ((size_t)bh * T_SZ + kbase + r) * HD_SZ + 2 * hh;
#pragma unroll
    for (int jn = 0; jn < 4; ++jn) {
      const float* Kpj = Kp + (size_t)jn * 16 * HD_SZ;
#pragma unroll
      for (int s = 0; s < 16; ++s) {
        v2f kb = *(const v2f*)(Kpj + 4 * s);
        c[jn] = wmma4(aq[s], kb, c[jn]);
      }
    }

    // ---- scale + causal mask (only diagonal block needs masking) ----
    const bool domask = (j == jmax);
#pragma unroll
    for (int jn = 0; jn < 4; ++jn)
#pragma unroll
      for (int v = 0; v < 8; ++v) {
        float sv = c[jn][v] * 0.125f;  // 1/sqrt(64)
        if (domask) {
          const int kcol = kbase + 16 * jn + r;
          const int qrow = q0 + v + 8 * hh;
          if (kcol > qrow) sv = -1.0e30f;
        }
        c[jn][v] = sv;
      }

    // ---- online softmax row stats (butterflies stay inside lane halves) ----
#pragma unroll
    for (int v = 0; v < 8; ++v) {
      float tmx = fmaxf(fmaxf(c[0][v], c[1][v]), fmaxf(c[2][v], c[3][v]));
      for (int off = 1; off < 16; off <<= 1)
        tmx = fmaxf(tmx, __shfl_xor(tmx, off, 32));
      const float mnew = fmaxf(mrow[v], tmx);
      const float alpha = __expf(mrow[v] - mnew);
      mrow[v] = mnew;
      float rs = 0.0f;
#pragma unroll
      for (int jn = 0; jn < 4; ++jn) {
        const float p = __expf(c[jn][v] - mnew);
        c[jn][v] = p;
        rs += p;
      }
      for (int off = 1; off < 16; off <<= 1) rs += __shfl_xor(rs, off, 32);
      lrow[v] = lrow[v] * alpha + rs;
#pragma unroll
      for (int jn = 0; jn < 4; ++jn) o[jn][v] *= alpha;
    }

    // ---- P: C-layout -> LDS row-major [16][64] ----
#pragma unroll
    for (int jn = 0; jn < 4; ++jn)
#pragma unroll
      for (int v = 0; v < 8; ++v)
        Pl[(v + 8 * hh) * 64 + 16 * jn + r] = c[jn][v];
    __syncthreads();

    // ---- O += P @ V ----
    const float* Vp =
        Vt + (size_t)bh * HD_SZ * T_SZ + (size_t)r * T_SZ + kbase + 2 * hh;
#pragma unroll
    for (int s2 = 0; s2 < 16; ++s2) {
      v2f ap = *(const v2f*)&Pl[r * 64 + 4 * s2 + 2 * hh];
#pragma unroll
      for (int jn = 0; jn < 4; ++jn) {
        v2f bv = *(const v2f*)(Vp + (size_t)jn * 16 * T_SZ + 4 * s2);
        o[jn] = wmma4(ap, bv, o[jn]);
      }
    }
    __syncthreads();
  }

  // ---- normalize + store into y[B,T,C] layout ----
  const int b = bh >> 4;
  const int head = bh & 15;
#pragma unroll
  for (int v = 0; v < 8; ++v) {
    const float inv = 1.0f / lrow[v];
    const int trow = q0 + v + 8 * hh;
    float* Yp = Y + (size_t)(b * T_SZ + trow) * C_SZ + head * HD_SZ + r;
#pragma unroll
    for (int jn = 0; jn < 4; ++jn) Yp[16 * jn] = o[jn][v] * inv;
  }
}

// ---------------------------------------------------------------------------
extern "C" void kernel_launch(void* const* d_in, const int* in_sizes, int n_in,
                              void* d_out, int out_size, void* d_ws,
                              size_t ws_size, hipStream_t stream) {
  const float* x = (const float*)d_in[0];
  const float* w_attn = (const float*)d_in[1];
  const float* b_attn = (const float*)d_in[2];
  const float* w_proj = (const float*)d_in[3];
  const float* b_proj = (const float*)d_in[4];
  float* out = (float*)d_out;

  const size_t BT = (size_t)B_SZ * T_SZ;           // 4096
  const size_t BHTD = (size_t)B_SZ * C_SZ * T_SZ;  // 4,194,304

  float* ws = (float*)d_ws;
  float* qkv = ws;                  // [4096, 3072]
  float* Qr = qkv + BT * 3 * C_SZ;  // [BH, T, 64]
  float* Kr = Qr + BHTD;            // [BH, T, 64]
  float* Vt = Kr + BHTD;            // [BH, 64, T]
  float* y = Vt + BHTD;             // [B, T, C]
  float* wAT = y + BT * C_SZ;       // w_attn^T  [3072, 1024]
  float* wPT = wAT + (size_t)3 * C_SZ * C_SZ;  // w_proj^T [1024, 1024]
  // total ws use: 33,554,432 floats = 128 MB

  // 0) one-time weight transposes (so GEMM B-fragments are contiguous b64)
  transpose_mat<<<(C_SZ / 32) * (3 * C_SZ / 32), 256, 0, stream>>>(
      w_attn, wAT, C_SZ, 3 * C_SZ);
  transpose_mat<<<(C_SZ / 32) * (C_SZ / 32), 256, 0, stream>>>(w_proj, wPT,
                                                               C_SZ, C_SZ);

  // 1) qkv = x @ w_attn + b_attn     (M=4096, N=3072, K=1024)
  gemm_bias_wmma<<<(BT / 64) * (3 * C_SZ / 64), 32, 0, stream>>>(
      x, wAT, b_attn, qkv, (int)BT, 3 * C_SZ, C_SZ);

  // 2) RoPE + layout reorg
  rope_reorg<<<(B_SZ * H_N * T_SZ) / 8, 256, 0, stream>>>(qkv, Qr, Kr, Vt);

  // 3) causal flash attention -> y[B,T,C]
  flash_attn<<<B_SZ * H_N * (T_SZ / 16), 32, 0, stream>>>(Qr, Kr, Vt, y);

  // 4) out = y @ w_proj + b_proj     (M=4096, N=1024, K=1024)
  gemm_bias_wmma<<<(BT / 64) * (C_SZ / 64), 32, 0, stream>>>(
      y, wPT, b_proj, out, (int)BT, C_SZ, C_SZ);
}